// CrossModalGSPFilter_67113158967396
// MI455X (gfx1250) — compile-verified
//
#include <hip/hip_runtime.h>
#include <hip/hip_bf16.h>

// Problem constants (reference: B=2, H=W=64, N=4096)
#define BB    2
#define NN    4096          // H*W
#define NT    256           // 16x16 tiles per matrix dimension
#define CG_T  384           // CG iterations (fixed -> deterministic graph)

typedef __attribute__((ext_vector_type(2))) float v2f;
typedef __attribute__((ext_vector_type(8))) float v8f;

// f = [px/(sqrt2*SIGMA_S), py/(sqrt2*SIGMA_S), rgb/(sqrt2*SIGMA_C)]
#define INV_S 7.0710678f    // 1/(sqrt(2)*0.1)
#define INV_C 14.1421356f   // 1/(sqrt(2)*0.05)
#define ALPHA 0.1f

// ---------------------------------------------------------------------------
// K1: build padded feature matrix F (B,N,8), squared norms (B,N), zero scalars
// ---------------------------------------------------------------------------
__global__ void k_features(const float* __restrict__ rgb,
                           float* __restrict__ F, float* __restrict__ nrm,
                           float* __restrict__ scal, int scalCount) {
    int g = blockIdx.x * 256 + threadIdx.x;        // 0 .. B*N-1
    int b = g >> 12, n = g & (NN - 1);
    int h = n >> 6, w = n & 63;
    const float* rg = rgb + (size_t)b * 3 * NN;
    float px = (float)w * (1.0f / 63.0f) * INV_S;
    float py = (float)h * (1.0f / 63.0f) * INV_S;
    float cr = rg[0 * NN + n] * INV_C;
    float cg = rg[1 * NN + n] * INV_C;
    float cb = rg[2 * NN + n] * INV_C;
    float* f = F + (size_t)g * 8;
    f[0] = px; f[1] = py; f[2] = cr; f[3] = cg; f[4] = cb;
    f[5] = 0.0f; f[6] = 0.0f; f[7] = 0.0f;
    nrm[g] = px * px + py * py + cr * cr + cg * cg + cb * cb;
    if (g < scalCount) scal[g] = 0.0f;             // rz slots
}

// ---------------------------------------------------------------------------
// K2: A = -ALPHA * W (off-diag), 0 on diag.  One 16x16 tile per wave32.
// Gram F*F^T via two V_WMMA_F32_16X16X4_F32 (K padded 5->8).
// ---------------------------------------------------------------------------
__global__ void k_buildA(const float* __restrict__ F,
                         const float* __restrict__ nrm,
                         float* __restrict__ A) {
    int wave = threadIdx.x >> 5;
    int lane = threadIdx.x & 31;
    int tile = blockIdx.x * 8 + wave;              // 0 .. B*NT*NT-1
    int b  = tile >> 16;
    int rm = tile & 65535;
    int ti = rm >> 8, tj = rm & 255;

    const float* Fb = F   + (size_t)b * NN * 8;
    const float* nb = nrm + (size_t)b * NN;

    int half = lane >> 4;                          // 0: K=0..1 / 1: K=2..3 (per VGPR pair)
    int l16  = lane & 15;
    int row  = ti * 16 + l16;
    int col  = tj * 16 + l16;
    int kb   = half * 2;

    v2f a0, b0, a1, b1;
    a0.x = Fb[row * 8 + kb];     a0.y = Fb[row * 8 + kb + 1];
    a1.x = Fb[row * 8 + kb + 4]; a1.y = Fb[row * 8 + kb + 5];
    b0.x = Fb[col * 8 + kb];     b0.y = Fb[col * 8 + kb + 1];
    b1.x = Fb[col * 8 + kb + 4]; b1.y = Fb[col * 8 + kb + 5];

    v8f c = {};
    c = __builtin_amdgcn_wmma_f32_16x16x4_f32(false, a0, false, b0, (short)0, c, false, false);
    c = __builtin_amdgcn_wmma_f32_16x16x4_f32(false, a1, false, b1, (short)0, c, false, false);

    float nj = nb[col];
    float* Ab = A + (size_t)b * NN * NN;
#pragma unroll
    for (int rr = 0; rr < 8; ++rr) {
        int i = ti * 16 + rr + half * 8;           // C/D layout: vgpr rr, halves
        int j = col;
        float wv = __expf(2.0f * c[rr] - nb[i] - nj);
        float val = (i == j) ? 0.0f : -ALPHA * wv;
        Ab[(size_t)i * NN + j] = val;
    }
}

// ---------------------------------------------------------------------------
// K3: per row i: A_ii = m_i - sum_j A_ij ; init b, x=0, r=b, z=r/diag, p=z.
// One row per wave32, 8 waves / block.  Wave-local fixed-order reduction.
// ---------------------------------------------------------------------------
__global__ void k_finalize(float* __restrict__ A, const float* __restrict__ depth,
                           float* __restrict__ diagA, float* __restrict__ xv,
                           float* __restrict__ rv, float* __restrict__ zv,
                           float* __restrict__ pv) {
    int wave = threadIdx.x >> 5, lane = threadIdx.x & 31;
    int rowg = blockIdx.x * 8 + wave;              // 0 .. B*N-1
    int b = rowg >> 12, i = rowg & (NN - 1);
    float* Arow = A + (size_t)b * NN * NN + (size_t)i * NN;
    const float4* Ar4 = (const float4*)Arow;
    float s = 0.0f;
    for (int it = lane; it < NN / 4; it += 32) {
        float4 v = Ar4[it];
        s += v.x + v.y + v.z + v.w;
    }
#pragma unroll
    for (int m = 16; m; m >>= 1) s += __shfl_xor(s, m, 32);
    if (lane == 0) {
        float y  = depth[rowg];
        float mi = (y != 0.0f) ? 1.0f : 0.0f;
        float d  = mi - s;                         // = m_i + ALPHA*deg_i
        Arow[i]      = d;
        diagA[rowg]  = d;
        float bb = mi * y;
        xv[rowg] = 0.0f;
        rv[rowg] = bb;
        float zz = bb / d;
        zv[rowg] = zz;
        pv[rowg] = zz;
    }
}

// ---------------------------------------------------------------------------
// K4: rz_0 = dot(r, z), one block per batch (deterministic tree reduction)
// ---------------------------------------------------------------------------
__global__ void k_rz0(const float* __restrict__ rv, const float* __restrict__ zv,
                      float* __restrict__ rzArr) {
    int b = blockIdx.x, tid = threadIdx.x;
    __shared__ float red[512];
    float s = 0.0f;
    for (int i = tid; i < NN; i += 512) s += rv[b * NN + i] * zv[b * NN + i];
    red[tid] = s; __syncthreads();
    for (int o = 256; o; o >>= 1) { if (tid < o) red[tid] += red[tid + o]; __syncthreads(); }
    if (tid == 0) rzArr[b * (CG_T + 1)] = red[0];
}

// ---------------------------------------------------------------------------
// K5: q = A * p.  One row per wave32 (float4 streamed from L2-resident A).
// ---------------------------------------------------------------------------
__global__ void k_matvec(const float* __restrict__ A, const float* __restrict__ pv,
                         float* __restrict__ qv) {
    int wave = threadIdx.x >> 5, lane = threadIdx.x & 31;
    int rowg = blockIdx.x * 8 + wave;
    int b = rowg >> 12, i = rowg & (NN - 1);
    const float4* Ar4 = (const float4*)(A + (size_t)b * NN * NN + (size_t)i * NN);
    const float4* p4  = (const float4*)(pv + b * NN);
    float s = 0.0f;
#pragma unroll 4
    for (int it = lane; it < NN / 4; it += 32) {
        float4 a = Ar4[it];
        float4 p = p4[it];
        s = fmaf(a.x, p.x, fmaf(a.y, p.y, fmaf(a.z, p.z, fmaf(a.w, p.w, s))));
    }
#pragma unroll
    for (int m = 16; m; m >>= 1) s += __shfl_xor(s, m, 32);
    if (lane == 0) qv[rowg] = s;
}

// ---------------------------------------------------------------------------
// K6: fused CG update, one block per batch.
//   pq = p.q ; alpha = rz_t/pq ; x += a p ; r -= a q ; z = r/diag ;
//   rz_{t+1} = r.z ; beta = rz_{t+1}/rz_t ; p = z + beta p
// ---------------------------------------------------------------------------
__global__ void k_cg_update(float* __restrict__ xv, float* __restrict__ rv,
                            float* __restrict__ zv, float* __restrict__ pv,
                            const float* __restrict__ qv, const float* __restrict__ diagA,
                            float* __restrict__ rzArr, int t) {
    int b = blockIdx.x, tid = threadIdx.x;
    const int base = b * NN;
    __shared__ float red[512];
    __shared__ float s_alpha, s_beta;

    float pq = 0.0f;
    for (int i = tid; i < NN; i += 512) pq += pv[base + i] * qv[base + i];
    red[tid] = pq; __syncthreads();
    for (int o = 256; o; o >>= 1) { if (tid < o) red[tid] += red[tid + o]; __syncthreads(); }
    if (tid == 0) {
        float rz = rzArr[b * (CG_T + 1) + t];
        float dn = red[0];
        s_alpha = (dn != 0.0f) ? rz / dn : 0.0f;
    }
    __syncthreads();
    float al = s_alpha;
    float rznew = 0.0f;
    for (int i = tid; i < NN; i += 512) {
        float pn = pv[base + i];
        float rn = rv[base + i] - al * qv[base + i];
        float zn = rn / diagA[base + i];
        xv[base + i] += al * pn;
        rv[base + i]  = rn;
        zv[base + i]  = zn;
        rznew += rn * zn;
    }
    __syncthreads();
    red[tid] = rznew; __syncthreads();
    for (int o = 256; o; o >>= 1) { if (tid < o) red[tid] += red[tid + o]; __syncthreads(); }
    if (tid == 0) {
        float rzold = rzArr[b * (CG_T + 1) + t];
        float rn = red[0];
        rzArr[b * (CG_T + 1) + t + 1] = rn;
        s_beta = (rzold != 0.0f) ? rn / rzold : 0.0f;
    }
    __syncthreads();
    float be = s_beta;
    for (int i = tid; i < NN; i += 512) pv[base + i] = zv[base + i] + be * pv[base + i];
}

// ---------------------------------------------------------------------------
// K7: out = relu(x)
// ---------------------------------------------------------------------------
__global__ void k_out(const float* __restrict__ xv, float* __restrict__ out) {
    int i = blockIdx.x * 256 + threadIdx.x;
    out[i] = fmaxf(xv[i], 0.0f);
}

// ---------------------------------------------------------------------------
extern "C" void kernel_launch(void* const* d_in, const int* in_sizes, int n_in,
                              void* d_out, int out_size, void* d_ws, size_t ws_size,
                              hipStream_t stream) {
    const float* rgb   = (const float*)d_in[0];    // (B,3,H,W)
    const float* depth = (const float*)d_in[1];    // (B,1,H,W)
    float* out = (float*)d_out;                    // (B,1,H,W)

    // Workspace layout (floats)
    float* ws = (float*)d_ws;
    size_t off = 0;
    float* A     = ws + off; off += (size_t)BB * NN * NN;   // 33,554,432
    float* F     = ws + off; off += (size_t)BB * NN * 8;
    float* nrm   = ws + off; off += (size_t)BB * NN;
    float* diagA = ws + off; off += (size_t)BB * NN;
    float* xv    = ws + off; off += (size_t)BB * NN;
    float* rv    = ws + off; off += (size_t)BB * NN;
    float* zv    = ws + off; off += (size_t)BB * NN;
    float* pv    = ws + off; off += (size_t)BB * NN;
    float* qv    = ws + off; off += (size_t)BB * NN;
    float* rzArr = ws + off; off += (size_t)BB * (CG_T + 1);
    (void)ws_size; (void)in_sizes; (void)n_in; (void)out_size;

    const int scalCount = BB * (CG_T + 1);

    // features + norms (+ zero rz slots every call: replay-safe)
    k_features<<<(BB * NN) / 256, 256, 0, stream>>>(rgb, F, nrm, rzArr, scalCount);

    // A = -ALPHA*W via WMMA tiles (8 tiles per 256-thread block)
    k_buildA<<<(BB * NT * NT) / 8, 256, 0, stream>>>(F, nrm, A);

    // diag, b, CG state init
    k_finalize<<<(BB * NN) / 8, 256, 0, stream>>>(A, depth, diagA, xv, rv, zv, pv);
    k_rz0<<<BB, 512, 0, stream>>>(rv, zv, rzArr);

    // Jacobi-preconditioned CG, fixed iteration count
    for (int t = 0; t < CG_T; ++t) {
        k_matvec<<<(BB * NN) / 8, 256, 0, stream>>>(A, pv, qv);
        k_cg_update<<<BB, 512, 0, stream>>>(xv, rv, zv, pv, qv, diagA, rzArr, t);
    }

    k_out<<<(BB * NN) / 256, 256, 0, stream>>>(xv, out);
}